// TokenMerger_44839458570826
// MI455X (gfx1250) — compile-verified
//
#include <hip/hip_runtime.h>
#include <hip/hip_bf16.h>

#define BATCH   32
#define NTOK    1024
#define DIM     768
#define SPLIT   512
#define RMERGE  307            // int(1024*0.3), < 1024-4
#define KEPTN   205            // SPLIT - RMERGE
#define OUTN    718            // 1 + KEPTN + SPLIT
#define TSTRIDE (16 * DIM)     // floats between consecutive 16-row dst tiles

typedef float v2f __attribute__((ext_vector_type(2)));
typedef float v8f __attribute__((ext_vector_type(8)));

// ---------------------------------------------------------------------------
// 1) reciprocal norms: rnorm[b*1024 + row] = 1 / max(||x||, 1e-12)
// ---------------------------------------------------------------------------
__global__ __launch_bounds__(256) void tm_rnorm(const float* __restrict__ tokens,
                                                float* __restrict__ rnorm) {
    const int row  = blockIdx.x * 8 + (threadIdx.x >> 5);
    const int lane = threadIdx.x & 31;
    const float* __restrict__ p = tokens + (size_t)row * DIM;
    float s = 0.0f;
    #pragma unroll 4
    for (int k = lane; k < DIM; k += 32) { float x = p[k]; s += x * x; }
    #pragma unroll
    for (int m = 16; m >= 1; m >>= 1) s += __shfl_xor(s, m, 32);
    if (lane == 0) rnorm[row] = 1.0f / fmaxf(sqrtf(s), 1e-12f);
}

// ---------------------------------------------------------------------------
// 2) fused similarity GEMM (fp32 WMMA) + row max/argmax
//    block = 4 waves, one (batch, 16-row src tile) per block.
//    wave w covers dst tiles [w*8, w*8+8), 4 tiles at a time with
//    4 independent accumulators and a software-pipelined k loop.
// ---------------------------------------------------------------------------
__global__ __launch_bounds__(128) void tm_sim(const float* __restrict__ tokens,
                                              const float* __restrict__ rnorm,
                                              float* __restrict__ scores,
                                              int* __restrict__ indices) {
    const int b    = blockIdx.x >> 5;        // 32 src tiles per batch
    const int tile = blockIdx.x & 31;
    const int wave = threadIdx.x >> 5;       // 0..3
    const int lane = threadIdx.x & 31;
    const int n16  = lane & 15;
    const int hi   = lane >> 4;

    const float* __restrict__ base   = tokens + (size_t)b * NTOK * DIM;
    const float* __restrict__ srcRow = base + (size_t)(tile * 16 + n16) * DIM + hi * 2;
    const float* __restrict__ rn     = rnorm + b * NTOK;

    float rs[8];
    #pragma unroll
    for (int r = 0; r < 8; ++r) rs[r] = rn[tile * 16 + r + 8 * hi];

    float bestV[8];
    int   bestI[8];
    #pragma unroll
    for (int r = 0; r < 8; ++r) { bestV[r] = -3.4e38f; bestI[r] = 0; }

    for (int g = 0; g < 2; ++g) {            // two groups of 4 dst tiles
        const int t0 = wave * 8 + g * 4;     // first dst tile of this group
        const float* __restrict__ dBase =
            base + (size_t)(SPLIT + t0 * 16 + n16) * DIM + hi * 2;

        if (g == 0) {                        // prefetch next group's B tiles
            const float* pf = dBase + 4 * TSTRIDE;
            __builtin_prefetch(pf, 0, 1);
            __builtin_prefetch(pf + TSTRIDE, 0, 1);
            __builtin_prefetch(pf + 2 * TSTRIDE, 0, 1);
            __builtin_prefetch(pf + 3 * TSTRIDE, 0, 1);
        }

        v8f acc[4];
        #pragma unroll
        for (int j = 0; j < 4; ++j) acc[j] = (v8f){};

        // software pipeline: current operands in regs, next chunk in flight
        v2f a = *(const v2f*)(srcRow);
        v2f bb[4];
        #pragma unroll
        for (int j = 0; j < 4; ++j) bb[j] = *(const v2f*)(dBase + j * TSTRIDE);

        #pragma unroll 2
        for (int k = 4; k < DIM; k += 4) {
            v2f an = *(const v2f*)(srcRow + k);
            v2f bn[4];
            #pragma unroll
            for (int j = 0; j < 4; ++j) bn[j] = *(const v2f*)(dBase + k + j * TSTRIDE);
            #pragma unroll
            for (int j = 0; j < 4; ++j)
                acc[j] = __builtin_amdgcn_wmma_f32_16x16x4_f32(
                             false, a, false, bb[j], (short)0, acc[j], false, false);
            a = an;
            #pragma unroll
            for (int j = 0; j < 4; ++j) bb[j] = bn[j];
        }
        #pragma unroll
        for (int j = 0; j < 4; ++j)
            acc[j] = __builtin_amdgcn_wmma_f32_16x16x4_f32(
                         false, a, false, bb[j], (short)0, acc[j], false, false);

        // scale by deferred normalization and update running max/argmax
        #pragma unroll
        for (int j = 0; j < 4; ++j) {
            const int   col = (t0 + j) * 16 + n16;
            const float rd  = rn[SPLIT + col];
            #pragma unroll
            for (int r = 0; r < 8; ++r) {
                float sim = acc[j][r] * rs[r] * rd;
                if (sim > bestV[r]) { bestV[r] = sim; bestI[r] = col; }
            }
        }
    }

    // intra-wave argmax reduce across the 16 column lanes
    #pragma unroll
    for (int m = 1; m < 16; m <<= 1) {
        #pragma unroll
        for (int r = 0; r < 8; ++r) {
            float ov = __shfl_xor(bestV[r], m, 32);
            int   oi = __shfl_xor(bestI[r], m, 32);
            if (ov > bestV[r] || (ov == bestV[r] && oi < bestI[r])) {
                bestV[r] = ov; bestI[r] = oi;
            }
        }
    }

    // cross-wave reduce through LDS: lv/li[wave][row]
    __shared__ float lv[64];
    __shared__ int   li[64];
    if (n16 == 0) {
        #pragma unroll
        for (int r = 0; r < 8; ++r) {
            lv[wave * 16 + r + 8 * hi] = bestV[r];
            li[wave * 16 + r + 8 * hi] = bestI[r];
        }
    }
    __syncthreads();
    if (threadIdx.x < 16) {
        const int row = threadIdx.x;
        float v = lv[row];
        int   i = li[row];
        #pragma unroll
        for (int w = 1; w < 4; ++w) {
            float ov = lv[w * 16 + row];
            int   oi = li[w * 16 + row];
            if (ov > v || (ov == v && oi < i)) { v = ov; i = oi; }
        }
        scores[b * SPLIT + tile * 16 + row]  = v;
        indices[b * SPLIT + tile * 16 + row] = i;
    }
}

// ---------------------------------------------------------------------------
// 3) top-k order + kept gather map via rank-by-counting (matches jax top_k:
//    descending, ties -> lower index first; kept list is stable/ascending).
//    Also materializes orderD = indices[order] for the parallel merge.
// ---------------------------------------------------------------------------
__global__ __launch_bounds__(SPLIT) void tm_rank(const float* __restrict__ scores,
                                                 const int* __restrict__ indices,
                                                 int* __restrict__ orderS,
                                                 int* __restrict__ orderD,
                                                 int* __restrict__ keptMap) {
    const int b = blockIdx.x;
    const int s = threadIdx.x;
    __shared__ float sc[SPLIT];
    __shared__ int   mflag[SPLIT];
    const float my = scores[b * SPLIT + s];
    sc[s] = my;
    __syncthreads();
    int rank = 0;
    for (int t = 0; t < SPLIT; ++t) {
        const float o = sc[t];
        rank += (o > my) || (o == my && t < s);
    }
    mflag[s] = (rank < RMERGE) ? 1 : 0;
    if (rank < RMERGE) {
        orderS[b * RMERGE + rank] = s;
        orderD[b * RMERGE + rank] = indices[b * SPLIT + s];
    }
    __syncthreads();
    if (rank >= RMERGE) {
        int before = 0;
        for (int t = 0; t < s; ++t) before += mflag[t];
        keptMap[b * KEPTN + (s - before)] = s;   // stable kept ordering
    }
}

// ---------------------------------------------------------------------------
// 4) assemble cls + kept-src rows of the output (dst region written by tm_merge)
// ---------------------------------------------------------------------------
__global__ __launch_bounds__(192) void tm_assemble(const float* __restrict__ tokens,
                                                   const float* __restrict__ cls,
                                                   const int* __restrict__ keptMap,
                                                   float* __restrict__ out) {
    const int b   = blockIdx.x / (1 + KEPTN);
    const int row = blockIdx.x % (1 + KEPTN);
    const float* src;
    if (row == 0) {
        src = cls + (size_t)b * DIM;
    } else {
        src = tokens + ((size_t)b * NTOK + keptMap[b * KEPTN + row - 1]) * DIM;
    }
    float* dstp = out + ((size_t)b * OUTN + row) * DIM;
    ((float4*)dstp)[threadIdx.x] = ((const float4*)src)[threadIdx.x];
}

// ---------------------------------------------------------------------------
// 5) parallel merge: one block per (batch, dst row). Replays, in global
//    top-k order, only the merges that target this row — per-thread
//    sequential (v+s)*0.5 reproduces the reference scan bitwise.
//    Writes the merged dst region of the output directly.
// ---------------------------------------------------------------------------
__global__ __launch_bounds__(192) void tm_merge(const float* __restrict__ tokens,
                                                const int* __restrict__ orderS,
                                                const int* __restrict__ orderD,
                                                float* __restrict__ out) {
    const int b = blockIdx.x >> 9;           // /512
    const int d = blockIdx.x & 511;
    const int tid = threadIdx.x;

    const float4* __restrict__ src4 =
        (const float4*)(tokens + (size_t)b * NTOK * DIM);
    float4 v = src4[(size_t)(SPLIT + d) * (DIM / 4) + tid];   // original dst row

    const int* __restrict__ oS = orderS + b * RMERGE;
    const int* __restrict__ oD = orderD + b * RMERGE;
    for (int i = 0; i < RMERGE; ++i) {
        if (oD[i] == d) {                    // uniform branch per block
            const float4 sv = src4[(size_t)oS[i] * (DIM / 4) + tid];
            v.x = (v.x + sv.x) * 0.5f;
            v.y = (v.y + sv.y) * 0.5f;
            v.z = (v.z + sv.z) * 0.5f;
            v.w = (v.w + sv.w) * 0.5f;
        }
    }
    ((float4*)out)[((size_t)b * OUTN + 1 + KEPTN + d) * (DIM / 4) + tid] = v;
}

// ---------------------------------------------------------------------------
extern "C" void kernel_launch(void* const* d_in, const int* in_sizes, int n_in,
                              void* d_out, int out_size, void* d_ws, size_t ws_size,
                              hipStream_t stream) {
    const float* tokens = (const float*)d_in[0];   // (32,1024,768) f32
    const float* cls    = (const float*)d_in[1];   // (32,1,768)    f32
    float* out = (float*)d_out;                    // (32,718,768)  f32

    char* ws = (char*)d_ws;
    float* rnorm   = (float*)(ws);                 // 32*1024*4 = 131072 B
    float* scores  = (float*)(ws + 131072);        // 32*512*4  =  65536 B
    int*   indices = (int*)  (ws + 196608);        // 32*512*4  =  65536 B
    int*   orderS  = (int*)  (ws + 262144);        // 32*307*4  =  39296 B
    int*   orderD  = (int*)  (ws + 301440);        // 32*307*4  =  39296 B
    int*   keptMap = (int*)  (ws + 340736);        // 32*205*4  =  26240 B

    tm_rnorm   <<<BATCH * NTOK / 8,      256, 0, stream>>>(tokens, rnorm);
    tm_sim     <<<BATCH * (SPLIT / 16),  128, 0, stream>>>(tokens, rnorm, scores, indices);
    tm_rank    <<<BATCH,               SPLIT, 0, stream>>>(scores, indices, orderS, orderD, keptMap);
    tm_assemble<<<BATCH * (1 + KEPTN),   192, 0, stream>>>(tokens, cls, keptMap, out);
    tm_merge   <<<BATCH * SPLIT,         192, 0, stream>>>(tokens, orderS, orderD, out);
}